// MultiHeadAttention_16673063043289
// MI455X (gfx1250) — compile-verified
//
#include <hip/hip_runtime.h>
#include <math.h>

// ---------------------------------------------------------------------------
// MHA forward for B=4, S=2048, D=512, H=8 (PHD=64) on gfx1250 (CDNA5, wave32)
// Precision-preserving path: V_WMMA_F32_16X16X4_F32 everywhere (problem is
// HBM-bound: the 537MB attn write dominates; f32 matrix pipe is ample).
// ---------------------------------------------------------------------------

typedef __attribute__((ext_vector_type(2))) float v2f;
typedef __attribute__((ext_vector_type(8))) float v8f;

#define DD    512
#define PHD   64
#define GG    32        // B*H groups after the torch-faithful flat reshape
#define LL    2048      // rows per group
#define MTOT  8192      // B*S
#define ROWS  32        // query rows per attention workgroup
#define LDSP  2052      // padded LDS row stride (floats): conflict-free WMMA A loads
#define SCALE 0.04419417382415922f  // 1/sqrt(512) (model-dim scale, as in source)
#define LN_EPS 1e-9f

__device__ __forceinline__ v8f wmma4(v2f a, v2f b, v8f c) {
  // D = A(16x4 f32) x B(4x16 f32) + C(16x16 f32)
  return __builtin_amdgcn_wmma_f32_16x16x4_f32(false, a, false, b, (short)0, c,
                                               false, false);
}

// ---------------------------------------------------------------------------
// out[M,N] = X[M,512] * W[N,512]^T + bias[N]     (M=8192, N=512)
// block = 8 waves; wave computes one 16x16 tile; block tile = 32x64
// ---------------------------------------------------------------------------
__global__ __launch_bounds__(256) void proj_kernel(const float* __restrict__ X,
                                                   const float* __restrict__ W,
                                                   const float* __restrict__ bias,
                                                   float* __restrict__ out) {
  const int wave = threadIdx.x >> 5;
  const int lane = threadIdx.x & 31;
  const int m15  = lane & 15;
  const int hi   = lane >> 4;
  const int koff = hi * 2;
  const int row0 = blockIdx.x * 32 + (wave >> 2) * 16;
  const int col0 = blockIdx.y * 64 + (wave & 3) * 16;

  v8f acc = {};
  const float* xrow = X + (size_t)(row0 + m15) * DD;
  const float* wrow = W + (size_t)(col0 + m15) * DD;
  for (int kb = 0; kb < DD; kb += 4) {
    v2f a = *(const v2f*)(xrow + kb + koff);
    v2f b = *(const v2f*)(wrow + kb + koff);   // B[k][n] = W[n][k]
    acc = wmma4(a, b, acc);
  }
  const float bc = bias[col0 + m15];
#pragma unroll
  for (int j = 0; j < 8; ++j)
    out[(size_t)(row0 + 8 * hi + j) * DD + col0 + m15] = acc[j] + bc;
}

// ---------------------------------------------------------------------------
// Attention for one (group g, 32-row stripe):
//   S = scale * Q Kt  -> LDS (32 x 2048, padded)
//   softmax rows in LDS, single coalesced HBM write of attn
//   C = P * V         -> ctx
// Dynamic LDS: 32*2052 + 32*8 + 32 floats (~257KB of the 320KB WGP LDS)
// ---------------------------------------------------------------------------
__global__ __launch_bounds__(256) void attn_kernel(const float* __restrict__ Q,
                                                   const float* __restrict__ K,
                                                   const float* __restrict__ V,
                                                   float* __restrict__ attn,
                                                   float* __restrict__ ctx) {
  extern __shared__ float sm[];
  float* sc   = sm;                    // ROWS x LDSP scores
  float* red  = sm + ROWS * LDSP;      // 32 x 8 partials
  float* rowv = red + ROWS * 8;        // 32 row max / row sum

  const int g    = blockIdx.y;
  const int rb   = blockIdx.x;
  const int tid  = threadIdx.x;
  const int wave = tid >> 5;
  const int lane = tid & 31;
  const int m15  = lane & 15;
  const int hi   = lane >> 4;
  const int koff = hi * 2;

  const float* Qg = Q + (size_t)g * LL * PHD;
  const float* Kg = K + (size_t)g * LL * PHD;
  const float* Vg = V + (size_t)g * LL * PHD;
  const int qrow0 = rb * ROWS;

  // ---- Phase 1: scores into LDS (2 x 128 tiles of 16x16, 8 waves) ----
  for (int t = wave; t < 256; t += 8) {
    const int tr = t >> 7;
    const int tc = t & 127;
    const int m0 = qrow0 + tr * 16;
    const int n0 = tc * 16;
    v8f acc = {};
    const float* qrow = Qg + (size_t)(m0 + m15) * PHD;
    const float* krow = Kg + (size_t)(n0 + m15) * PHD;
    for (int pb = 0; pb < PHD; pb += 4) {
      v2f a = *(const v2f*)(qrow + pb + koff);
      v2f b = *(const v2f*)(krow + pb + koff); // B[p][n] = K[n][p]
      acc = wmma4(a, b, acc);
    }
#pragma unroll
    for (int j = 0; j < 8; ++j)
      sc[(tr * 16 + 8 * hi + j) * LDSP + n0 + m15] = acc[j] * SCALE;
  }
  __syncthreads();

  // ---- Phase 2: row softmax in LDS (8 threads per row, 256 cols each) ----
  const int r   = tid >> 3;
  const int seg = tid & 7;
  float* srow = sc + r * LDSP + seg * 256;

  float mx = -3.4028235e38f;
  for (int i = 0; i < 256; ++i) mx = fmaxf(mx, srow[i]);
  red[r * 8 + seg] = mx;
  __syncthreads();
  if (seg == 0) {
    float m = red[r * 8];
    for (int i = 1; i < 8; ++i) m = fmaxf(m, red[r * 8 + i]);
    rowv[r] = m;
  }
  __syncthreads();
  const float rmax = rowv[r];
  float s = 0.f;
  for (int i = 0; i < 256; ++i) {
    const float e = __expf(srow[i] - rmax);
    srow[i] = e;
    s += e;
  }
  red[r * 8 + seg] = s;
  __syncthreads();
  if (seg == 0) {
    float t = 0.f;
    for (int i = 0; i < 8; ++i) t += red[r * 8 + i];
    rowv[r] = t;
  }
  __syncthreads();
  const float inv = 1.0f / rowv[r];
  for (int i = 0; i < 256; ++i) srow[i] *= inv;
  __syncthreads();

  // single coalesced write of the normalized 32x2048 stripe
  float* arow = attn + (size_t)g * LL * LL + (size_t)qrow0 * LL;
  for (int idx = tid; idx < ROWS * LL; idx += 256) {
    const int rr = idx >> 11;
    const int cc = idx & 2047;
    arow[(size_t)rr * LL + cc] = sc[rr * LDSP + cc];
  }
  __syncthreads();

  // ---- Phase 3: context C[32x64] = P[32x2048] * V[2048x64] ----
  {
    const int tr = wave >> 2;   // 0..1
    const int tc = wave & 3;    // 0..3
    v8f acc = {};
    const float* prow = sc + (tr * 16 + m15) * LDSP;   // A from LDS (padded)
    const float* vcol = Vg + tc * 16 + m15;
    for (int nb = 0; nb < LL; nb += 4) {
      v2f a = *(const v2f*)(prow + nb + koff);
      v2f b;
      b.x = vcol[(size_t)(nb + koff) * PHD];           // B[k][d] = V[k][d]
      b.y = vcol[(size_t)(nb + koff + 1) * PHD];
      acc = wmma4(a, b, acc);
    }
#pragma unroll
    for (int j = 0; j < 8; ++j)
      ctx[(size_t)g * LL * PHD +
          (size_t)(qrow0 + tr * 16 + 8 * hi + j) * PHD + tc * 16 + m15] = acc[j];
  }
}

// ---------------------------------------------------------------------------
// out = LayerNorm(resid + ctx * Wo^T + bo)   (16 rows per block, fused)
// ---------------------------------------------------------------------------
__global__ __launch_bounds__(256) void outln_kernel(const float* __restrict__ ctx,
                                                    const float* __restrict__ Wo,
                                                    const float* __restrict__ bo,
                                                    const float* __restrict__ resid,
                                                    const float* __restrict__ wa,
                                                    const float* __restrict__ wb,
                                                    float* __restrict__ out) {
  __shared__ float xb[16 * DD];      // 32KB row staging
  __shared__ float red[16 * 16];
  __shared__ float rowm[16];
  __shared__ float rowsd[16];

  const int tid  = threadIdx.x;
  const int wave = tid >> 5;
  const int lane = tid & 31;
  const int m15  = lane & 15;
  const int hi   = lane >> 4;
  const int koff = hi * 2;
  const int row0 = blockIdx.x * 16;

  for (int t = wave; t < 32; t += 8) {
    const int col0 = t * 16;
    v8f acc = {};
    const float* arow = ctx + (size_t)(row0 + m15) * DD;
    const float* wrow = Wo + (size_t)(col0 + m15) * DD;
    for (int kb = 0; kb < DD; kb += 4) {
      v2f a = *(const v2f*)(arow + kb + koff);
      v2f b = *(const v2f*)(wrow + kb + koff);
      acc = wmma4(a, b, acc);
    }
    const int cc = col0 + m15;
    const float bc = bo[cc];
#pragma unroll
    for (int j = 0; j < 8; ++j) {
      const int rr = 8 * hi + j;
      xb[rr * DD + cc] = acc[j] + bc + resid[(size_t)(row0 + rr) * DD + cc];
    }
  }
  __syncthreads();

  // LayerNorm: 16 threads per row, 32 cols each; unbiased std (ddof=1),
  // eps added to std (torch variant replicated by the reference)
  const int r   = tid >> 4;
  const int seg = tid & 15;
  const float* x = xb + r * DD + seg * 32;

  float s = 0.f;
  for (int i = 0; i < 32; ++i) s += x[i];
  red[r * 16 + seg] = s;
  __syncthreads();
  if (seg == 0) {
    float t = 0.f;
    for (int i = 0; i < 16; ++i) t += red[r * 16 + i];
    rowm[r] = t * (1.0f / 512.0f);
  }
  __syncthreads();
  const float mean = rowm[r];
  float vs = 0.f;
  for (int i = 0; i < 32; ++i) {
    const float d = x[i] - mean;
    vs += d * d;
  }
  red[r * 16 + seg] = vs;
  __syncthreads();
  if (seg == 0) {
    float t = 0.f;
    for (int i = 0; i < 16; ++i) t += red[r * 16 + i];
    rowsd[r] = sqrtf(t * (1.0f / 511.0f));
  }
  __syncthreads();
  const float istd = 1.0f / (rowsd[r] + LN_EPS);
  for (int i = 0; i < 32; ++i) {
    const int c = seg * 32 + i;
    out[(size_t)(row0 + r) * DD + c] = wa[c] * (x[i] - mean) * istd + wb[c];
  }
}

// ---------------------------------------------------------------------------
extern "C" void kernel_launch(void* const* d_in, const int* in_sizes, int n_in,
                              void* d_out, int out_size, void* d_ws, size_t ws_size,
                              hipStream_t stream) {
  const float* key   = (const float*)d_in[0];
  const float* value = (const float*)d_in[1];
  const float* query = (const float*)d_in[2];
  const float* Wq    = (const float*)d_in[3];
  const float* bq    = (const float*)d_in[4];
  const float* Wk    = (const float*)d_in[5];
  const float* bk    = (const float*)d_in[6];
  const float* Wv    = (const float*)d_in[7];
  const float* bv    = (const float*)d_in[8];
  const float* Wo    = (const float*)d_in[9];
  const float* bo    = (const float*)d_in[10];
  const float* wa    = (const float*)d_in[11];
  const float* wb    = (const float*)d_in[12];

  float* out  = (float*)d_out;                  // [8192,512]
  float* attn = out + (size_t)MTOT * DD;        // [32,2048,2048]

  float* qb = (float*)d_ws;                     // 4 x 16MB scratch
  float* kb = qb + (size_t)MTOT * DD;
  float* vb = kb + (size_t)MTOT * DD;
  float* cb = vb + (size_t)MTOT * DD;

  dim3 gproj(MTOT / 32, DD / 64);
  proj_kernel<<<gproj, 256, 0, stream>>>(query, Wq, bq, qb);
  proj_kernel<<<gproj, 256, 0, stream>>>(key,   Wk, bk, kb);
  proj_kernel<<<gproj, 256, 0, stream>>>(value, Wv, bv, vb);

  const size_t attn_lds =
      (size_t)(ROWS * LDSP + ROWS * 8 + ROWS) * sizeof(float);
  hipFuncSetAttribute((const void*)attn_kernel,
                      hipFuncAttributeMaxDynamicSharedMemorySize,
                      (int)attn_lds);
  attn_kernel<<<dim3(LL / ROWS, GG), 256, attn_lds, stream>>>(qb, kb, vb, attn, cb);

  outln_kernel<<<MTOT / 16, 256, 0, stream>>>(cb, Wo, bo, query, wa, wb, out);
}